// message_lstm_84834194031116
// MI455X (gfx1250) — compile-verified
//
#include <hip/hip_runtime.h>
#include <cstdint>

#define LSTM_B 32
#define LSTM_T 512
#define LSTM_I 512
#define LSTM_H 1024
#define LSTM_G (4 * LSTM_H) /* 4096 */
#define NWG 64
#define NTHREADS 256

typedef __attribute__((ext_vector_type(2))) float v2f;
typedef __attribute__((ext_vector_type(8))) float v8f;

__device__ __forceinline__ float fast_sig(float v) {
    return 1.0f / (1.0f + __expf(-v));
}
// tanh(x) = 1 - 2/(exp(2x)+1): overflow-safe (+/-inf exp -> +/-1, no NaN)
__device__ __forceinline__ float fast_tanh(float v) {
    return 1.0f - 2.0f / (__expf(2.0f * v) + 1.0f);
}

// Pack W[G=4H rows][K cols] (row-major) into pair-interleaved B-operand layout:
//   WP[(k>>1)*(2*G) + col*2 + (k&1)] = W[col*K + k]
// so a lane's two K-values for a WMMA B fragment are one contiguous b64.
__global__ void pack_pairs(const float* __restrict__ src, float* __restrict__ dst,
                           int K) {
    long i = (long)blockIdx.x * blockDim.x + threadIdx.x;
    long n = (long)LSTM_G * K;
    if (i >= n) return;
    int col = (int)(i / K); // 0..4H-1
    int k   = (int)(i % K);
    dst[(size_t)(k >> 1) * (2 * LSTM_G) + (size_t)col * 2 + (k & 1)] = src[i];
}

__global__ void bias_init(const float* __restrict__ bx, const float* __restrict__ bh,
                          float* __restrict__ bsum) {
    int i = blockIdx.x * blockDim.x + threadIdx.x;
    if (i < LSTM_G) bsum[i] = bx[i] + bh[i];
}

// zero both h ping-pong buffers (contiguous) and the barrier counters
__global__ void state_init(float* __restrict__ hbufs, unsigned* __restrict__ bar) {
    int i = blockIdx.x * blockDim.x + threadIdx.x;
    if (i < 2 * LSTM_B * LSTM_H) hbufs[i] = 0.0f;
    if (i == 0) { bar[0] = 0u; bar[1] = 0u; }
}

// Fully parallel phase: xg[m][col] = bias[col] + sum_k x[m][k]*Wx[col][k]
// m = b*T + t  (x is [B,T,I] row-major => row m directly). One 16x16 tile per wave.
__global__ __launch_bounds__(NTHREADS)
void xg_gemm(const float* __restrict__ x, const float* __restrict__ WxP,
             const float* __restrict__ bsum, float* __restrict__ xg) {
    const int tid  = threadIdx.x;
    const int wave = tid >> 5;
    const int lane = tid & 31;
    const int half = lane >> 4;
    const int n16  = lane & 15;
    const int mt   = blockIdx.x;            // 0..(B*T/16)-1 = 0..1023
    const int nt   = blockIdx.y * 8 + wave; // 0..255
    const int col  = nt * 16 + n16;
    const int m    = mt * 16 + n16;         // A row supplied by this lane

    const float bias = bsum[col];
    const float* __restrict__ xa = x + (size_t)m * LSTM_I + half * 2;
    const float* __restrict__ bw = WxP + ((size_t)half * LSTM_G + col) * 2;

    v8f acc = {bias, bias, bias, bias, bias, bias, bias, bias};
#pragma unroll 8
    for (int k = 0; k < LSTM_I; k += 4) {
        v2f a = *(const v2f*)(xa + k);
        v2f b = *(const v2f*)(bw + (size_t)k * LSTM_G);
        acc = __builtin_amdgcn_wmma_f32_16x16x4_f32(false, a, false, b,
                                                    (short)0, acc, false, false);
    }
#pragma unroll
    for (int r = 0; r < 8; ++r)
        xg[(size_t)(mt * 16 + r + half * 8) * LSTM_G + col] = acc[r];
}

__global__ __launch_bounds__(NTHREADS, 1)
void lstm_persistent(const float* __restrict__ x, const float* __restrict__ WxP,
                     const float* __restrict__ WhP, const float* __restrict__ bsum,
                     const float* __restrict__ xg, /* may be null -> fused x GEMM */
                     float* __restrict__ hA, float* __restrict__ hB,
                     float* __restrict__ out, unsigned* __restrict__ bar) {
    __shared__ float smem[6 * 256]; // 6 gate tiles (f,g,o for each of 2 batch tiles)

    const int tid  = threadIdx.x;
    const int wave = tid >> 5;
    const int lane = tid & 31;
    const int half = lane >> 4;   // 0: lanes 0-15, 1: lanes 16-31
    const int n16  = lane & 15;   // column-within-tile / A-row-within-tile
    const int jt   = blockIdx.x;  // 0..63  (H column tile)
    const int bt   = wave >> 2;   // 0..1   (batch tile)
    const int gate = wave & 3;    // 0=i 1=f 2=g 3=o
    const int col  = gate * LSTM_H + jt * 16 + n16; // column in [0,4H)
    const int arow = bt * 16 + n16;                 // batch row this lane supplies for A

    const float bias = bsum[col];

    const float* __restrict__ bwx = WxP + ((size_t)half * LSTM_G + col) * 2;
    const float* __restrict__ bwh = WhP + ((size_t)half * LSTM_G + col) * 2;

    // Per-lane xg gather offsets for the C/D layout: VGPR r -> row bt*16 + r + half*8
    size_t xgoff[8];
    if (xg) {
#pragma unroll
        for (int r = 0; r < 8; ++r)
            xgoff[r] = ((size_t)(bt * 16 + r + half * 8) * LSTM_T) * LSTM_G + col;
    }

    v8f c = {0.f, 0.f, 0.f, 0.f, 0.f, 0.f, 0.f, 0.f}; // cell state lives in registers

    v8f pre; // software-pipelined xg tile for the upcoming step
    if (xg) {
#pragma unroll
        for (int r = 0; r < 8; ++r) pre[r] = xg[xgoff[r]]; // t = 0
    }

    for (int t = 0; t < LSTM_T; ++t) {
        const float* __restrict__ hread  = (t & 1) ? hB : hA;
        float* __restrict__       hwrite = (t & 1) ? hA : hB;

        v8f acc;
        if (xg) {
            acc = pre; // bias already folded into xg
        } else {
            acc = (v8f){bias, bias, bias, bias, bias, bias, bias, bias};
            // ---- fused input projection: acc += x[b,t,:] @ Wx^T[:,col] ----
            const float* __restrict__ xa =
                x + ((size_t)arow * LSTM_T + t) * LSTM_I + half * 2;
#pragma unroll 8
            for (int k = 0; k < LSTM_I; k += 4) {
                v2f a = *(const v2f*)(xa + k);
                v2f b = *(const v2f*)(bwx + (size_t)k * LSTM_G);
                acc = __builtin_amdgcn_wmma_f32_16x16x4_f32(false, a, false, b,
                                                            (short)0, acc, false, false);
            }
        }

        // ---- recurrent slice: acc += h[b,:] @ Wh^T[:,col] ----
        const float* __restrict__ ha = hread + (size_t)arow * LSTM_H + half * 2;
#pragma unroll 8
        for (int k = 0; k < LSTM_H; k += 4) {
            v2f a = *(const v2f*)(ha + k);
            v2f b = *(const v2f*)(bwh + (size_t)k * LSTM_G);
            acc = __builtin_amdgcn_wmma_f32_16x16x4_f32(false, a, false, b,
                                                        (short)0, acc, false, false);
        }

        // ---- exchange f/g/o gate tiles through LDS; gate-i wave combines ----
        if (gate != 0) {
            float* s = &smem[(bt * 3 + (gate - 1)) * 256];
#pragma unroll
            for (int r = 0; r < 8; ++r) s[r * 32 + lane] = acc[r];
        }
        __syncthreads();
        if (gate == 0) {
            const float* sf = &smem[(bt * 3 + 0) * 256];
            const float* sg = &smem[(bt * 3 + 1) * 256];
            const float* so = &smem[(bt * 3 + 2) * 256];
#pragma unroll
            for (int r = 0; r < 8; ++r) {
                float iv = fast_sig(acc[r]);
                float fv = fast_sig(sf[r * 32 + lane]);
                float gv = fast_tanh(sg[r * 32 + lane]);
                float ov = fast_sig(so[r * 32 + lane]);
                float cv = fv * c[r] + iv * gv;
                c[r] = cv;
                float hv = ov * fast_tanh(cv);
                // C/D layout: VGPR r -> M = r + half*8, N = lane&15
                int brow = bt * 16 + r + half * 8;
                int jc   = jt * 16 + n16;
                hwrite[(size_t)brow * LSTM_H + jc] = hv;
                out[((size_t)brow * LSTM_T + t) * LSTM_H + jc] = hv;
            }
        }

        // ---- prefetch next step's xg tile; its latency hides in the barrier ----
        if (xg && t + 1 < LSTM_T) {
            const float* xgt = xg + (size_t)(t + 1) * LSTM_G;
#pragma unroll
            for (int r = 0; r < 8; ++r) pre[r] = xgt[xgoff[r]];
        }

        // ---- device-wide barrier (arrive + generation) ----
        __syncthreads();
        if (tid == 0) {
            __threadfence();
            unsigned old = __hip_atomic_load(&bar[1], __ATOMIC_ACQUIRE,
                                             __HIP_MEMORY_SCOPE_AGENT);
            unsigned a = __hip_atomic_fetch_add(&bar[0], 1u, __ATOMIC_ACQ_REL,
                                                __HIP_MEMORY_SCOPE_AGENT);
            if (a == NWG - 1) {
                __hip_atomic_store(&bar[0], 0u, __ATOMIC_RELAXED,
                                   __HIP_MEMORY_SCOPE_AGENT);
                __hip_atomic_fetch_add(&bar[1], 1u, __ATOMIC_ACQ_REL,
                                       __HIP_MEMORY_SCOPE_AGENT);
            } else {
                while (__hip_atomic_load(&bar[1], __ATOMIC_ACQUIRE,
                                         __HIP_MEMORY_SCOPE_AGENT) == old) {
                    __builtin_amdgcn_s_sleep(4);
                }
            }
        }
        __syncthreads();
    }
}

extern "C" void kernel_launch(void* const* d_in, const int* in_sizes, int n_in,
                              void* d_out, int out_size, void* d_ws, size_t ws_size,
                              hipStream_t stream) {
    (void)in_sizes; (void)n_in; (void)out_size;

    const float* x  = (const float*)d_in[0];
    const float* Wx = (const float*)d_in[1];
    const float* bx = (const float*)d_in[2];
    const float* Wh = (const float*)d_in[3];
    const float* bh = (const float*)d_in[4];
    float* out = (float*)d_out;

    char* ws = (char*)d_ws;
    size_t off = 0;
    float* WxP = (float*)(ws + off); off += (size_t)LSTM_I * LSTM_G * sizeof(float); // 8 MB
    float* WhP = (float*)(ws + off); off += (size_t)LSTM_H * LSTM_G * sizeof(float); // 16 MB
    float* bsum = (float*)(ws + off); off += (size_t)LSTM_G * sizeof(float);
    float* hA = (float*)(ws + off);  off += (size_t)LSTM_B * LSTM_H * sizeof(float);
    float* hB = (float*)(ws + off);  off += (size_t)LSTM_B * LSTM_H * sizeof(float);
    unsigned* bar = (unsigned*)(ws + off); off += 256; // padded/aligned
    float* xg = (float*)(ws + off);
    size_t need_xg = off + (size_t)LSTM_B * LSTM_T * LSTM_G * sizeof(float); // +256 MB
    const bool use_xg = (ws_size >= need_xg);

    {
        long n = (long)LSTM_G * LSTM_I;
        pack_pairs<<<(unsigned)((n + 255) / 256), 256, 0, stream>>>(Wx, WxP, LSTM_I);
    }
    {
        long n = (long)LSTM_G * LSTM_H;
        pack_pairs<<<(unsigned)((n + 255) / 256), 256, 0, stream>>>(Wh, WhP, LSTM_H);
    }
    bias_init<<<(LSTM_G + 255) / 256, 256, 0, stream>>>(bx, bh, bsum);
    state_init<<<(2 * LSTM_B * LSTM_H + 255) / 256, 256, 0, stream>>>(hA, bar);

    if (use_xg) {
        dim3 grid(LSTM_B * LSTM_T / 16, LSTM_G / (8 * 16)); // (1024, 32)
        xg_gemm<<<grid, NTHREADS, 0, stream>>>(x, WxP, bsum, xg);
    }

    lstm_persistent<<<NWG, NTHREADS, 0, stream>>>(x, WxP, WhP, bsum,
                                                  use_xg ? xg : nullptr,
                                                  hA, hB, out, bar);
}